// GAT_Layer_60215441490521
// MI455X (gfx1250) — compile-verified
//
#include <hip/hip_runtime.h>
#include <hip/hip_bf16.h>
#include <cstdint>

// ---------------------------------------------------------------------------
// GAT layer for MI455X (gfx1250, wave32, WMMA).
// Memory-bound workload (~1 GB traffic -> ~45us floor @ 23.3 TB/s).
// GEMM (100k x 64 x 64) done with V_WMMA_F32_16X16X4_F32 in f32 to match
// reference precision; scatter phase relies on f32 global atomics hitting
// the 192MB L2 (all atomic targets < 30MB).
// Round-1 refinement: W kept TRANSPOSED in LDS so both A and B fragments are
// contiguous 8B-aligned 2-float LDS loads -> ds_load_b64 straight into the
// even-aligned WMMA operand pairs (removes the v_mov shuffle churn seen in
// round 0's B-fragment path).
// ---------------------------------------------------------------------------

typedef __attribute__((ext_vector_type(2))) float v2f;
typedef __attribute__((ext_vector_type(8))) float v8f;

#define DFEAT 64
#define NS_GAT 0.2f
#define NS_OUT 0.01f

__device__ __forceinline__ float lrelu(float v, float ns) {
  return v > 0.0f ? v : v * ns;
}

// Order-preserving float -> uint mapping so segment-max can use atomicMax(u32).
// 0u is below the image of every finite/inf float, so it is the identity init.
__device__ __forceinline__ unsigned fmap(float f) {
  unsigned u = __float_as_uint(f);
  return (u & 0x80000000u) ? ~u : (u | 0x80000000u);
}
__device__ __forceinline__ float funmap(unsigned k) {
  return (k & 0x80000000u) ? __uint_as_float(k ^ 0x80000000u)
                           : __uint_as_float(~k);
}

// ---------------------------------------------------------------------------
// Kernel 1: embedding gather + h = emb[x] @ W (WMMA f32 16x16x4) + attention
// logits a_src/a_dst via in-register fragment reduction.
// Block = 256 threads = 8 waves, each wave owns a 16-row stripe (128 rows/blk).
// ---------------------------------------------------------------------------
__global__ __launch_bounds__(256)
void gat_node_kernel(const int* __restrict__ x,
                     const float* __restrict__ emb,
                     const float* __restrict__ Wm,
                     const float* __restrict__ att_src,
                     const float* __restrict__ att_dst,
                     float* __restrict__ h,
                     float* __restrict__ a_src,
                     float* __restrict__ a_dst,
                     int N) {
  // W transposed: sWt[n][k] = W[k][n]; stride 68 -> bank = (4n + k) mod 64,
  // conflict-free for the 32-lane fragment reads (halves offset by 2 mod 4).
  __shared__ float sWt[DFEAT][68];        // ~17.4 KB
  __shared__ float sA[8][16][68];         // per-wave gathered tile, same pad
  __shared__ float sAs[DFEAT], sAd[DFEAT];

  const int tid  = threadIdx.x;
  const int wave = tid >> 5;
  const int lane = tid & 31;
  const int half = lane >> 4;   // K-half for A/B fragments, M-half for C/D
  const int lr   = lane & 15;

  for (int i = tid; i < DFEAT * DFEAT; i += 256) {
    int k = i >> 6, n = i & 63;
    sWt[n][k] = Wm[i];                    // coalesced read, transposed store
  }
  if (tid < DFEAT) { sAs[tid] = att_src[tid]; sAd[tid] = att_dst[tid]; }
  __syncthreads();

  const int row_base = blockIdx.x * 128 + wave * 16;

  // Gather 16 embedding rows into LDS; each row is a coalesced 256B read
  // split across the wave (lanes read cols lane and lane+32).
  for (int rr = 0; rr < 16; ++rr) {
    int grow = row_base + rr;
    int node = x[grow < N ? grow : 0];      // clamp tail rows, stores guarded
    const float* rp = emb + (size_t)node * DFEAT;
    sA[wave][rr][lane]      = rp[lane];
    sA[wave][rr][lane + 32] = rp[lane + 32];
  }
  // same-wave LDS RAW handled by compiler-inserted s_wait_dscnt

  v8f acc[4];
#pragma unroll
  for (int nt = 0; nt < 4; ++nt)
#pragma unroll
    for (int i = 0; i < 8; ++i) acc[nt][i] = 0.0f;

  // f32 WMMA layout (ISA 7.12.2): A 16x4 -> lanes 0-15 carry K={k0,k0+1},
  // lanes 16-31 carry K={k0+2,k0+3}; B 4x16 rows striped across lanes with
  // the same half pairing. Folding "+2*half" into k0 covers both operands.
  // k0 is always even -> the 2-float fragment loads are 8B aligned b64 loads.
  for (int kk = 0; kk < 16; ++kk) {
    const int k0 = kk * 4 + 2 * half;
    v2f a = *(const v2f*)&sA[wave][lr][k0];
#pragma unroll
    for (int nt = 0; nt < 4; ++nt) {
      v2f b = *(const v2f*)&sWt[nt * 16 + lr][k0];
      acc[nt] = __builtin_amdgcn_wmma_f32_16x16x4_f32(
          /*neg_a=*/false, a, /*neg_b=*/false, b,
          /*c_mod=*/(short)0, acc[nt], /*reuse_a=*/false, /*reuse_b=*/false);
    }
  }

  // C/D layout: lane<16 VGPR r = D[M=r][N=lr], lane>=16 = D[M=r+8][N=lr].
#pragma unroll
  for (int nt = 0; nt < 4; ++nt)
#pragma unroll
    for (int r = 0; r < 8; ++r) {
      int grow = row_base + r + 8 * half;
      if (grow < N) h[(size_t)grow * DFEAT + nt * 16 + lr] = acc[nt][r];
    }

  // a_src/a_dst: dot each produced row with attention vectors, reducing the
  // N dimension (striped over 16 lanes per half) with shfl_xor.
  float ps[8], pd[8];
#pragma unroll
  for (int r = 0; r < 8; ++r) { ps[r] = 0.0f; pd[r] = 0.0f; }
#pragma unroll
  for (int nt = 0; nt < 4; ++nt) {
    float as = sAs[nt * 16 + lr];
    float ad = sAd[nt * 16 + lr];
#pragma unroll
    for (int r = 0; r < 8; ++r) {
      ps[r] += acc[nt][r] * as;
      pd[r] += acc[nt][r] * ad;
    }
  }
#pragma unroll
  for (int off = 1; off < 16; off <<= 1) {
#pragma unroll
    for (int r = 0; r < 8; ++r) {
      ps[r] += __shfl_xor(ps[r], off, 32);
      pd[r] += __shfl_xor(pd[r], off, 32);
    }
  }
  if (lr == 0) {
#pragma unroll
    for (int r = 0; r < 8; ++r) {
      int grow = row_base + r + 8 * half;
      if (grow < N) { a_src[grow] = ps[r]; a_dst[grow] = pd[r]; }
    }
  }
}

// ---------------------------------------------------------------------------
// Kernel 0: deterministic state init (harness does not re-poison buffers).
// ---------------------------------------------------------------------------
__global__ void gat_init_kernel(float* __restrict__ out,
                                unsigned* __restrict__ m_bits,
                                float* __restrict__ s, int N) {
  int t = blockIdx.x * blockDim.x + threadIdx.x;
  if (t < N * DFEAT) out[t] = 0.0f;       // x_gat accumulator region of d_out
  if (t < N) { m_bits[t] = 0u; s[t] = 0.0f; }
}

// ---------------------------------------------------------------------------
// Kernel 2: per-edge logit + segment max (uint-mapped atomicMax, L2 resident).
// Edges E..E+N-1 are the implicit self-loops.
// ---------------------------------------------------------------------------
__global__ void gat_edge_max_kernel(const int* __restrict__ ei,
                                    const float* __restrict__ a_src,
                                    const float* __restrict__ a_dst,
                                    float* __restrict__ e,
                                    unsigned* __restrict__ m_bits,
                                    int E, int N) {
  int t = blockIdx.x * blockDim.x + threadIdx.x;
  if (t >= E + N) return;
  int src, dst;
  if (t < E) { src = ei[t]; dst = ei[E + t]; } else { src = dst = t - E; }
  float ev = lrelu(a_src[src] + a_dst[dst], NS_GAT);
  e[t] = ev;
  atomicMax(&m_bits[dst], fmap(ev));
}

// ---------------------------------------------------------------------------
// Kernel 3: segment sum of exp(e - m).
// ---------------------------------------------------------------------------
__global__ void gat_edge_sum_kernel(const int* __restrict__ ei,
                                    const float* __restrict__ e,
                                    const unsigned* __restrict__ m_bits,
                                    float* __restrict__ s,
                                    int E, int N) {
  int t = blockIdx.x * blockDim.x + threadIdx.x;
  if (t >= E + N) return;
  int dst = (t < E) ? ei[E + t] : (t - E);
  float ex = expf(e[t] - funmap(m_bits[dst]));
  atomicAdd(&s[dst], ex);
}

// ---------------------------------------------------------------------------
// Kernel 4: alpha + weighted scatter. 64 threads per edge -> coalesced reads
// of h[src] and coalesced f32 atomics into the L2-resident output tile.
// ---------------------------------------------------------------------------
__global__ void gat_scatter_kernel(const int* __restrict__ ei,
                                   const float* __restrict__ e,
                                   const unsigned* __restrict__ m_bits,
                                   const float* __restrict__ s,
                                   const float* __restrict__ h,
                                   float* __restrict__ out,
                                   int E, int N) {
  int t = blockIdx.x * blockDim.x + threadIdx.x;
  int edge = t >> 6;
  int c = t & 63;
  if (edge >= E + N) return;
  int src, dst;
  if (edge < E) { src = ei[edge]; dst = ei[E + edge]; } else { src = dst = edge - E; }
  float alpha = expf(e[edge] - funmap(m_bits[dst])) / (s[dst] + 1e-16f);
  if (c == 0) out[(size_t)N * DFEAT + edge] = alpha;   // second tuple output
  atomicAdd(&out[(size_t)dst * DFEAT + c],
            h[(size_t)src * DFEAT + c] * alpha);
}

// ---------------------------------------------------------------------------
// Kernel 5: bias + outer leaky_relu, in place.
// ---------------------------------------------------------------------------
__global__ void gat_final_kernel(float* __restrict__ out,
                                 const float* __restrict__ bias, int N) {
  int t = blockIdx.x * blockDim.x + threadIdx.x;
  if (t >= N * DFEAT) return;
  out[t] = lrelu(out[t] + bias[t & 63], NS_OUT);
}

// ---------------------------------------------------------------------------
extern "C" void kernel_launch(void* const* d_in, const int* in_sizes, int n_in,
                              void* d_out, int out_size, void* d_ws, size_t ws_size,
                              hipStream_t stream) {
  (void)n_in; (void)out_size; (void)ws_size;
  const int*   x    = (const int*)d_in[0];
  const int*   ei   = (const int*)d_in[1];   // [2,E] row-major: src then dst
  const float* emb  = (const float*)d_in[2];
  const float* Wm   = (const float*)d_in[3];
  const float* asv  = (const float*)d_in[4];
  const float* adv  = (const float*)d_in[5];
  const float* bias = (const float*)d_in[6];
  const int N = in_sizes[0];
  const int E = in_sizes[1] / 2;
  const int EN = E + N;
  float* out = (float*)d_out;                // [N*64] out | [E+N] alpha

  // workspace layout (~34 MB)
  float* ws      = (float*)d_ws;
  float* h       = ws;                               // N*64
  float* a_src   = h + (size_t)N * DFEAT;            // N
  float* a_dst   = a_src + N;                        // N
  float* e       = a_dst + N;                        // E+N
  unsigned* m_b  = (unsigned*)(e + (size_t)EN);      // N
  float* s       = (float*)(m_b + N);                // N

  gat_init_kernel<<<(N * DFEAT + 255) / 256, 256, 0, stream>>>(out, m_b, s, N);
  gat_node_kernel<<<(N + 127) / 128, 256, 0, stream>>>(
      x, emb, Wm, asv, adv, h, a_src, a_dst, N);
  gat_edge_max_kernel<<<(EN + 255) / 256, 256, 0, stream>>>(
      ei, a_src, a_dst, e, m_b, E, N);
  gat_edge_sum_kernel<<<(EN + 255) / 256, 256, 0, stream>>>(
      ei, e, m_b, s, E, N);
  long total = (long)EN * DFEAT;
  gat_scatter_kernel<<<(int)((total + 255) / 256), 256, 0, stream>>>(
      ei, e, m_b, s, h, out, E, N);
  gat_final_kernel<<<(N * DFEAT + 255) / 256, 256, 0, stream>>>(out, bias, N);
}